// ExiGCNLayerLoRA_19782619365924
// MI455X (gfx1250) — compile-verified
//
#include <hip/hip_runtime.h>

#define NN     50000
#define DD     256
#define OO     256
#define RANK   8
#define SCAL   2.0f      // LORA_ALPHA / RANK = 16/8
#define EDGES  800000
#define DEDGES 80000

typedef float v2f __attribute__((ext_vector_type(2)));
typedef float v8f __attribute__((ext_vector_type(8)));

// ---------------- zero acc1 (ws) and B region of d_out ----------------
__global__ void zero2_kernel(float* __restrict__ a, float* __restrict__ b, long n) {
    long i = (long)blockIdx.x * blockDim.x + threadIdx.x;
    long stride = (long)gridDim.x * blockDim.x;
    for (; i < n; i += stride) { a[i] = 0.0f; b[i] = 0.0f; }
}

// ------- adj SpMM: acc1 += val * dF[col]  ;  Bout += val * F[col] -------
// one wave per edge, 8 floats per lane (2x float4 gather, 16 f32 atomics)
__global__ void spmm_adj_kernel(const int* __restrict__ row, const int* __restrict__ col,
                                const float* __restrict__ val,
                                const float* __restrict__ F, const float* __restrict__ dF,
                                float* __restrict__ acc1, float* __restrict__ Bout, int nE) {
    int gid  = blockIdx.x * blockDim.x + threadIdx.x;
    int e    = gid >> 5;
    int lane = gid & 31;
    if (e >= nE) return;
    int   r = row[e];
    int   c = col[e];
    float v = val[e];
    const float4* fc  = (const float4*)(F  + (long)c * DD) + lane * 2;
    const float4* dfc = (const float4*)(dF + (long)c * DD) + lane * 2;
    float4 f0 = fc[0],  f1 = fc[1];
    float4 g0 = dfc[0], g1 = dfc[1];
    float* a1 = acc1 + (long)r * DD + lane * 8;
    float* bo = Bout + (long)r * DD + lane * 8;
    unsafeAtomicAdd(a1 + 0, v * g0.x); unsafeAtomicAdd(a1 + 1, v * g0.y);
    unsafeAtomicAdd(a1 + 2, v * g0.z); unsafeAtomicAdd(a1 + 3, v * g0.w);
    unsafeAtomicAdd(a1 + 4, v * g1.x); unsafeAtomicAdd(a1 + 5, v * g1.y);
    unsafeAtomicAdd(a1 + 6, v * g1.z); unsafeAtomicAdd(a1 + 7, v * g1.w);
    unsafeAtomicAdd(bo + 0, v * f0.x); unsafeAtomicAdd(bo + 1, v * f0.y);
    unsafeAtomicAdd(bo + 2, v * f0.z); unsafeAtomicAdd(bo + 3, v * f0.w);
    unsafeAtomicAdd(bo + 4, v * f1.x); unsafeAtomicAdd(bo + 5, v * f1.y);
    unsafeAtomicAdd(bo + 6, v * f1.z); unsafeAtomicAdd(bo + 7, v * f1.w);
}

// ------- dadj SpMM: acc1 += val * (F[col] + dF[col])  (fused two SpMMs) -------
__global__ void spmm_dadj_kernel(const int* __restrict__ row, const int* __restrict__ col,
                                 const float* __restrict__ val,
                                 const float* __restrict__ F, const float* __restrict__ dF,
                                 float* __restrict__ acc1, int nE) {
    int gid  = blockIdx.x * blockDim.x + threadIdx.x;
    int e    = gid >> 5;
    int lane = gid & 31;
    if (e >= nE) return;
    int   r = row[e];
    int   c = col[e];
    float v = val[e];
    const float4* fc  = (const float4*)(F  + (long)c * DD) + lane * 2;
    const float4* dfc = (const float4*)(dF + (long)c * DD) + lane * 2;
    float4 f0 = fc[0],  f1 = fc[1];
    float4 g0 = dfc[0], g1 = dfc[1];
    float* a1 = acc1 + (long)r * DD + lane * 8;
    unsafeAtomicAdd(a1 + 0, v * (f0.x + g0.x)); unsafeAtomicAdd(a1 + 1, v * (f0.y + g0.y));
    unsafeAtomicAdd(a1 + 2, v * (f0.z + g0.z)); unsafeAtomicAdd(a1 + 3, v * (f0.w + g0.w));
    unsafeAtomicAdd(a1 + 4, v * (f1.x + g1.x)); unsafeAtomicAdd(a1 + 5, v * (f1.y + g1.y));
    unsafeAtomicAdd(a1 + 6, v * (f1.z + g1.z)); unsafeAtomicAdd(a1 + 7, v * (f1.w + g1.w));
}

// ---------------- Bout += acc1  (B = adj@F + F_input) ----------------
__global__ void add_b_kernel(float* __restrict__ Bout, const float* __restrict__ acc1, long n) {
    long i = (long)blockIdx.x * blockDim.x + threadIdx.x;
    long stride = (long)gridDim.x * blockDim.x;
    for (; i < n; i += stride) Bout[i] += acc1[i];
}

// ---------------- delta_W = (lora_A @ lora_B) * SCALING ----------------
__global__ void lora_kernel(const float* __restrict__ A, const float* __restrict__ B,
                            float* __restrict__ dW) {
    int d = blockIdx.x;   // 0..255
    int o = threadIdx.x;  // 0..255
    float s = 0.0f;
#pragma unroll
    for (int r = 0; r < RANK; ++r) s += A[d * RANK + r] * B[r * OO + o];
    dW[d * OO + o] = s * SCAL;
}

// ---------------- f32 WMMA GEMM: D = (Cin ? Cin : 0) + A[M,K] @ B[K,OO] ----------------
// block = 128 threads (4 waves); each wave owns one 16x16 tile, block covers 16x64
// grid = (OO/64, M/16).  Uses V_WMMA_F32_16X16X4_F32, 64 WMMAs per tile (K=256).
__global__ void gemm_wmma_f32_kernel(const float* __restrict__ A,
                                     const float* __restrict__ Bmat,
                                     const float* __restrict__ Cin,
                                     float* __restrict__ Dout, int K) {
    int wave = threadIdx.x >> 5;
    int lane = threadIdx.x & 31;
    int m0   = blockIdx.y * 16;
    int n0   = blockIdx.x * 64 + wave * 16;
    int half = lane >> 4;   // selects K pair (A/B) / upper M half (C,D)
    int l16  = lane & 15;

    v8f c = {};
    if (Cin) {
#pragma unroll
        for (int i = 0; i < 8; ++i)
            c[i] = Cin[(long)(m0 + i + half * 8) * OO + n0 + l16];
    }

    // A fragment base: lane l16 = M row, half*2 = starting K of this lane's pair
    const float* arow = A + (long)(m0 + l16) * K + half * 2;
    // B fragment base: lane l16 = N col, half*2 = starting K row
    const float* bcol = Bmat + (long)(half * 2) * OO + n0 + l16;

#pragma unroll 8
    for (int kk = 0; kk < K; kk += 4) {
        v2f a, b;
        a.x = arow[kk];
        a.y = arow[kk + 1];
        b.x = bcol[(long)kk * OO];
        b.y = bcol[(long)(kk + 1) * OO];
        // (neg_a, A, neg_b, B, c_mod, C, reuse_a, reuse_b)
        c = __builtin_amdgcn_wmma_f32_16x16x4_f32(false, a, false, b, (short)0, c,
                                                  false, false);
    }

#pragma unroll
    for (int i = 0; i < 8; ++i)
        Dout[(long)(m0 + i + half * 8) * OO + n0 + l16] = c[i];
}

extern "C" void kernel_launch(void* const* d_in, const int* in_sizes, int n_in,
                              void* d_out, int out_size, void* d_ws, size_t ws_size,
                              hipStream_t stream) {
    const float* features = (const float*)d_in[0];
    const float* dfeat    = (const float*)d_in[1];
    const int*   adj_row  = (const int*)d_in[2];
    const int*   adj_col  = (const int*)d_in[3];
    const float* adj_val  = (const float*)d_in[4];
    const int*   del_row  = (const int*)d_in[5];
    const int*   del_col  = (const int*)d_in[6];
    const float* del_val  = (const float*)d_in[7];
    const float* W        = (const float*)d_in[8];
    // d_in[9] = bias (all zeros, unused by reference math)
    const float* lora_A   = (const float*)d_in[10];
    const float* lora_B   = (const float*)d_in[11];

    const long nd = (long)NN * DD;                 // 12.8M floats
    float* out_Z     = (float*)d_out;              // new_Z    [N, OUT]
    float* out_fixed = out_Z + nd;                 // fixed    [N, OUT]
    float* out_B     = out_Z + 2 * nd;             // B        [N, D]

    float* acc1 = (float*)d_ws;                    // F_input accumulator [N, D]
    float* dW   = acc1 + nd;                       // delta_W [D, OUT]

    // 1) zero acc1 and B output region (atomic targets)
    zero2_kernel<<<1024, 256, 0, stream>>>(acc1, out_B, nd);

    // 2) SpMMs (one wave per edge; 8 edges per 256-thread block)
    spmm_adj_kernel<<<EDGES / 8, 256, 0, stream>>>(adj_row, adj_col, adj_val,
                                                   features, dfeat, acc1, out_B, EDGES);
    spmm_dadj_kernel<<<DEDGES / 8, 256, 0, stream>>>(del_row, del_col, del_val,
                                                     features, dfeat, acc1, DEDGES);

    // 3) B = adj@F + F_input
    add_b_kernel<<<1024, 256, 0, stream>>>(out_B, acc1, nd);

    // 4) delta_W = (lora_A @ lora_B) * 2.0
    lora_kernel<<<DD, OO, 0, stream>>>(lora_A, lora_B, dW);

    // 5) fixed = F_input @ W          (WMMA f32)
    dim3 ggrid(OO / 64, NN / 16);
    gemm_wmma_f32_kernel<<<ggrid, 128, 0, stream>>>(acc1, W, nullptr, out_fixed, DD);

    // 6) new_Z = fixed + B @ delta_W  (WMMA f32, C seeded from fixed)
    gemm_wmma_f32_kernel<<<ggrid, 128, 0, stream>>>(out_B, dW, out_fixed, out_Z, DD);
}